// DFFN_12902081757734
// MI455X (gfx1250) — compile-verified
//
#include <hip/hip_runtime.h>
#include <math.h>

typedef float v2f __attribute__((ext_vector_type(2)));
typedef float v8f __attribute__((ext_vector_type(8)));

#define B_   4
#define CIN  64
#define C2   256
#define H_   256
#define W_   256
#define HW   65536

// cos(2*pi*n/8)
__device__ __constant__ float c8tab[8] = {
    1.0f, 0.70710678118654752f, 0.0f, -0.70710678118654752f,
   -1.0f, -0.70710678118654752f, 0.0f, 0.70710678118654752f};

// Generic pointers into LDS carry the wave-relative LDS byte address in their
// low 32 bits (CDNA5 ISA 10.2) -> VDST operand for
// the async-to-LDS instructions.
__device__ __forceinline__ unsigned lds_addr32(const void* p) {
  return (unsigned)(size_t)p;
}

// Async DMA of one 16-byte chunk: copied from global address base + goff to the LDS address in loff, ASYNCcnt-tracked.
__device__ __forceinline__ void async_b128_to_lds(unsigned loff, unsigned goff,
                                                  const void* base) {
  asm volatile("global_load_async_to_lds_b128 %0, %1, %2"
               :: "v"(loff), "v"(goff), "s"(base)
               : "memory");
}

__device__ __forceinline__ void wait_asynccnt0() {
  asm volatile("s_wait_asynccnt 0x0" ::: "memory");
}

// ---------------------------------------------------------------------------
// Kernel 0: per-channel circular-conv kernel  k = irfft2(F)   (256 x 8 x 8)
// irfft2(rfft2(P)*F) == circ_conv2(P, irfft2(F)) for real F.
// k[p][q] = (1/64) * sum_{u,v} G[u][v] * cos(2pi(up+vq)/8),
// G = Hermitian extension of F (8x5 -> 8x8).
// ---------------------------------------------------------------------------
__global__ __launch_bounds__(256) void dffn_precompute_k(
    const float* __restrict__ f, float* __restrict__ ktab) {
  const int o = threadIdx.x;           // channel 0..255
  const float* fo = f + o * 40;        // (8,5) filter for this channel
  for (int p = 0; p < 8; ++p) {
    for (int q = 0; q < 8; ++q) {
      float acc = 0.f;
      for (int u = 0; u < 8; ++u) {
        for (int v = 0; v < 8; ++v) {
          const float g = (v <= 4) ? fo[u * 5 + v]
                                   : fo[((8 - u) & 7) * 5 + (8 - v)];
          acc += g * c8tab[(u * p + v * q) & 7];
        }
      }
      ktab[o * 64 + p * 8 + q] = acc * (1.0f / 64.0f);
    }
  }
}

// ---------------------------------------------------------------------------
// Kernel 1: project_in (1x1 conv, WMMA f32 GEMM) + bias + FFT filter
// (as per-channel 8x8 circular conv over each 64-contiguous-pixel segment).
// One workgroup per (b, row h). 256 threads = 8 waves.
// Dynamic LDS: ylds[256][65] + klds[256][65]  (stride 65 = conflict-free)
// klds is filled by async-to-LDS DMA overlapped with the first GEMM segment.
// ---------------------------------------------------------------------------
__global__ __launch_bounds__(256) void dffn_proj_fft(
    const float* __restrict__ x, const float* __restrict__ w_in,
    const float* __restrict__ b_in, const float* __restrict__ ktab,
    float* __restrict__ y) {
  extern __shared__ float smem[];
  float* ylds = smem;              // 256*65 floats
  float* klds = smem + 256 * 65;   // 256*65 floats

  const int t    = threadIdx.x;
  const int lane = t & 31;
  const int wave = t >> 5;
  const int blk  = blockIdx.x;     // 0..1023
  const int b    = blk >> 8;
  const int h    = blk & 255;

  // ---- async DMA: per-channel conv kernels -> LDS (16B chunks) ----
  // 256 channels x 16 chunks; row stride padded to 65 floats.
  for (int i = t; i < C2 * 16; i += 256) {
    const int o  = i >> 4;
    const int c4 = (i & 15) << 2;
    async_b128_to_lds(lds_addr32(&klds[o * 65 + c4]),
                      (unsigned)((o * 64 + c4) * 4), ktab);
  }

  const float bin = b_in[t];       // thread t owns channel t in conv phase
  const float* xb = x + (size_t)b * CIN * HW + h * W_;
  float*       yb = y + (size_t)b * C2 * HW + h * W_;

  const int mrow  = lane & 15;          // row/col within 16-wide tile
  const int ksel  = (lane >> 4) << 1;   // K offset 0 or 2 (A/B frag layout)
  const int mbase = (lane >> 4) << 3;   // D frag: upper lanes hold M+8

  float ksum = 0.f;  // sum of conv kernel (bias folded through linear filter)

  for (int seg = 0; seg < 4; ++seg) {
    const int w0 = seg << 6;

    // ---- GEMM: y[256 och][64 px] = w_in[256x64] @ x[64x64] ----
    // wave handles ch-tiles {wave*32, wave*32+16} x px-tiles {0,16,32,48}
    v8f acc[2][4];
#pragma unroll
    for (int i = 0; i < 2; ++i)
#pragma unroll
      for (int j = 0; j < 4; ++j) acc[i][j] = (v8f)0.0f;

    for (int k0 = 0; k0 < 64; k0 += 4) {
      const int kk = k0 + ksel;
      v2f a[2], bb[4];
#pragma unroll
      for (int i = 0; i < 2; ++i) {
        const int m = wave * 32 + i * 16 + mrow;
        a[i].x = w_in[m * 64 + kk];
        a[i].y = w_in[m * 64 + kk + 1];
      }
#pragma unroll
      for (int j = 0; j < 4; ++j) {
        const int px = w0 + j * 16 + mrow;
        bb[j].x = xb[(size_t)kk * HW + px];
        bb[j].y = xb[(size_t)(kk + 1) * HW + px];
      }
#pragma unroll
      for (int i = 0; i < 2; ++i)
#pragma unroll
        for (int j = 0; j < 4; ++j)
          acc[i][j] = __builtin_amdgcn_wmma_f32_16x16x4_f32(
              false, a[i], false, bb[j], (short)0, acc[i][j], false, false);
    }

    if (seg == 0) wait_asynccnt0();  // klds DMA retired (own wave)
    __syncthreads();                 // all waves: klds ready / ylds readers done

    // scatter D fragments to LDS: ylds[och][px]
#pragma unroll
    for (int i = 0; i < 2; ++i)
#pragma unroll
      for (int j = 0; j < 4; ++j)
#pragma unroll
        for (int r = 0; r < 8; ++r)
          ylds[(wave * 32 + i * 16 + mbase + r) * 65 + j * 16 + mrow] =
              acc[i][j][r];
    __syncthreads();

    // ---- FFT filter == circular conv, thread t = channel t ----
    {
      float* row = ylds + t * 65;
      const float* kr = klds + t * 65;
      if (seg == 0) {
#pragma unroll
        for (int s = 0; s < 64; ++s) ksum += kr[s];
      }
      float outv[64];
#pragma unroll
      for (int pq = 0; pq < 64; ++pq) {
        const int p = pq >> 3, q = pq & 7;
        float s = bin * ksum;  // bias folded through linear filter
        for (int pp = 0; pp < 8; ++pp) {
          const int krow = ((p - pp) & 7) << 3;
          for (int qq = 0; qq < 8; ++qq)
            s += row[pp * 8 + qq] * kr[krow + ((q - qq) & 7)];
        }
        outv[pq] = s;
      }
#pragma unroll
      for (int j = 0; j < 64; ++j) row[j] = outv[j];
    }
    __syncthreads();

    // coalesced store of the segment
    for (int i = t; i < C2 * 64; i += 256)
      yb[(size_t)(i >> 6) * HW + w0 + (i & 63)] = ylds[(i >> 6) * 65 + (i & 63)];
    __syncthreads();
  }
}

// ---------------------------------------------------------------------------
// Kernel 2: depthwise 3x3 + bias + GELU-gate + output 1x1 proj (WMMA) + bias
// One workgroup per (b, h, 64-px segment). Dynamic LDS:
//   wlds[64][129] (w_out, async-DMA'd), glds[128][65] (gated activations)
// ---------------------------------------------------------------------------
__global__ __launch_bounds__(256) void dffn_dw_gate_proj(
    const float* __restrict__ y, const float* __restrict__ w_dw,
    const float* __restrict__ b_dw, const float* __restrict__ w_out,
    const float* __restrict__ b_out, float* __restrict__ out) {
  extern __shared__ float smem[];
  float* wlds = smem;              // 64*129 floats
  float* glds = smem + 64 * 129;   // 128*65 floats

  const int t    = threadIdx.x;
  const int lane = t & 31;
  const int wave = t >> 5;
  const int blk  = blockIdx.x;     // 0..4095
  const int b    = blk >> 10;
  const int h    = (blk >> 2) & 255;
  const int w0   = (blk & 3) << 6;

  // ---- async DMA: w_out -> LDS (overlapped with depthwise/gate phase) ----
  // 64 rows x 32 chunks of 16B; row stride padded to 129 floats.
  for (int i = t; i < 64 * 32; i += 256) {
    const int fz = i >> 5;
    const int c4 = (i & 31) << 2;
    async_b128_to_lds(lds_addr32(&wlds[fz * 129 + c4]),
                      (unsigned)((fz * 128 + c4) * 4), w_out);
  }

  // ---- depthwise conv + GELU gate into LDS: glds[c][px], c in 0..127 ----
  const float* yb = y + (size_t)b * C2 * HW;
  for (int i = t; i < 128 * 64; i += 256) {
    const int c  = i >> 6;
    const int px = i & 63;
    const int w  = w0 + px;
    const int o2 = c + 128;
    // pre-touch upcoming channel rows (global_prefetch_b8)
    if (i + 256 < 128 * 64) {
      const int cn = (i + 256) >> 6;
      __builtin_prefetch(&yb[(size_t)cn * HW + h * W_ + w0 + ((i + 256) & 63)],
                         0, 0);
      __builtin_prefetch(
          &yb[(size_t)(cn + 128) * HW + h * W_ + w0 + ((i + 256) & 63)], 0, 0);
    }
    float t1 = b_dw[c];
    float t2 = b_dw[o2];
    const float* wd1 = w_dw + c * 9;
    const float* wd2 = w_dw + o2 * 9;
#pragma unroll
    for (int dy = -1; dy <= 1; ++dy) {
      const int hh = h + dy;
      if (hh < 0 || hh >= H_) continue;
#pragma unroll
      for (int dx = -1; dx <= 1; ++dx) {
        const int ww = w + dx;
        if (ww < 0 || ww >= W_) continue;
        const int ki = (dy + 1) * 3 + (dx + 1);
        t1 += yb[(size_t)c * HW + hh * W_ + ww] * wd1[ki];
        t2 += yb[(size_t)o2 * HW + hh * W_ + ww] * wd2[ki];
      }
    }
    const float g = 0.5f * t1 * (1.0f + erff(t1 * 0.70710678118654752f));
    glds[c * 65 + px] = g * t2;
  }
  wait_asynccnt0();   // w_out DMA retired (own wave)
  __syncthreads();    // all waves: wlds + glds ready

  // ---- GEMM: out[64 f][64 px] = w_out[64x128] @ g[128x64] ----
  const int fi    = wave >> 1;          // f-tile 0..3
  const int mrow  = lane & 15;
  const int ksel  = (lane >> 4) << 1;
  const int mbase = (lane >> 4) << 3;

  v8f acc[2];
  acc[0] = (v8f)0.0f;
  acc[1] = (v8f)0.0f;

  for (int k0 = 0; k0 < 128; k0 += 4) {
    const int kk = k0 + ksel;
    v2f a;
    a.x = wlds[(fi * 16 + mrow) * 129 + kk];
    a.y = wlds[(fi * 16 + mrow) * 129 + kk + 1];
#pragma unroll
    for (int jj = 0; jj < 2; ++jj) {
      const int n0 = (((wave & 1) << 1) + jj) << 4;
      v2f bb;
      bb.x = glds[kk * 65 + n0 + mrow];
      bb.y = glds[(kk + 1) * 65 + n0 + mrow];
      acc[jj] = __builtin_amdgcn_wmma_f32_16x16x4_f32(
          false, a, false, bb, (short)0, acc[jj], false, false);
    }
  }

  float* ob = out + (size_t)b * 64 * HW + h * W_ + w0;
#pragma unroll
  for (int jj = 0; jj < 2; ++jj) {
    const int n0 = (((wave & 1) << 1) + jj) << 4;
#pragma unroll
    for (int r = 0; r < 8; ++r) {
      const int fch = fi * 16 + mbase + r;
      ob[(size_t)fch * HW + n0 + mrow] = acc[jj][r] + b_out[fch];
    }
  }
}

// ---------------------------------------------------------------------------
extern "C" void kernel_launch(void* const* d_in, const int* in_sizes, int n_in,
                              void* d_out, int out_size, void* d_ws,
                              size_t ws_size, hipStream_t stream) {
  (void)in_sizes; (void)n_in; (void)out_size; (void)ws_size;
  const float* x     = (const float*)d_in[0];
  const float* w_in  = (const float*)d_in[1];
  const float* b_in  = (const float*)d_in[2];
  const float* fftf  = (const float*)d_in[3];
  const float* w_dw  = (const float*)d_in[4];
  const float* b_dw  = (const float*)d_in[5];
  const float* w_out = (const float*)d_in[6];
  const float* b_out = (const float*)d_in[7];
  float* out = (float*)d_out;

  float* ktab = (float*)d_ws;                          // 256*64 floats
  float* y    = (float*)((char*)d_ws + (64 * 1024));   // 4*256*256*256 floats

  const size_t smem1 = (size_t)(2 * 256 * 65) * sizeof(float);        // 133120
  const size_t smem2 = (size_t)(64 * 129 + 128 * 65) * sizeof(float); // 66304
  hipFuncSetAttribute((const void*)dffn_proj_fft,
                      hipFuncAttributeMaxDynamicSharedMemorySize, (int)smem1);
  hipFuncSetAttribute((const void*)dffn_dw_gate_proj,
                      hipFuncAttributeMaxDynamicSharedMemorySize, (int)smem2);

  dffn_precompute_k<<<1, 256, 0, stream>>>(fftf, ktab);
  dffn_proj_fft<<<dim3(B_ * H_), dim3(256), smem1, stream>>>(x, w_in, b_in,
                                                             ktab, y);
  dffn_dw_gate_proj<<<dim3(B_ * H_ * (W_ / 64)), dim3(256), smem2, stream>>>(
      y, w_dw, b_dw, w_out, b_out, out);
}